// MapEncoder_33500744909487
// MI455X (gfx1250) — compile-verified
//
#include <hip/hip_runtime.h>
#include <math.h>

#define NE 500000
#define NV 2048
#define DD 64
#define NH 4

typedef float v2f __attribute__((ext_vector_type(2)));
typedef float v8f __attribute__((ext_vector_type(8)));

// ---------- helpers ----------
__device__ __forceinline__ unsigned enc_f(float f) {
    unsigned u = __float_as_uint(f);
    return (u & 0x80000000u) ? ~u : (u | 0x80000000u);
}
__device__ __forceinline__ float dec_f(unsigned e) {
    unsigned u = (e & 0x80000000u) ? (e ^ 0x80000000u) : ~e;
    return __uint_as_float(u);
}
__device__ __forceinline__ float elu1(float x) {
    return x > 0.0f ? x : (__expf(x) - 1.0f);
}

// workspace layout (floats)
#define WS_NUM   0                       // NV*256
#define WS_DEN   (NV*256)                // NV*4
#define WS_TV    (WS_DEN + NV*4)         // NV*4
#define WS_Q     (WS_TV + NV*4)          // 4*64
#define WS_C     (WS_Q + 256)            // 4
#define WS_MX    (WS_C + 4)              // 4 (as unsigned)

// ---------- K_init: zero accumulators, init max ----------
__global__ void k_init(float* ws) {
    const int n = NV*256 + NV*4;
    for (int i = blockIdx.x * blockDim.x + threadIdx.x; i < n;
         i += gridDim.x * blockDim.x)
        ws[i] = 0.0f;
    int t = blockIdx.x * blockDim.x + threadIdx.x;
    if (t < 4) ((unsigned*)(ws + WS_MX))[t] = enc_f(-INFINITY);
}

// ---------- K0: tiny precompute ----------
// g[h] = Wh[h] @ a_src ; q[h] = W2 @ g[h] ; c[h] = b2 . g[h]
// u[h] = Wh[h] @ a_tgt ; tV[v][h] = v_enc[v] . u[h]
__global__ void k_pre(const float* __restrict__ Wh, const float* __restrict__ ah,
                      const float* __restrict__ W2, const float* __restrict__ b2,
                      const float* __restrict__ venc, float* ws) {
    __shared__ float gS[256], uS[256];
    int tid = threadIdx.x;
    int h = tid >> 6, i = tid & 63;
    float g = 0.f, u = 0.f;
    for (int j = 0; j < DD; ++j) {
        float w = Wh[h*4096 + i*64 + j];
        g += w * ah[h*130 + j];
        u += w * ah[h*130 + 64 + j];
    }
    gS[tid] = g; uS[tid] = u;
    __syncthreads();
    float q = 0.f;
    for (int j = 0; j < DD; ++j) q += W2[i*64 + j] * gS[h*64 + j];
    ws[WS_Q + tid] = q;
    if (tid < 4) {
        float c = 0.f;
        for (int j = 0; j < DD; ++j) c += b2[j] * gS[tid*64 + j];
        ws[WS_C + tid] = c;
    }
    for (int v = tid; v < NV; v += blockDim.x) {
        for (int hh = 0; hh < NH; ++hh) {
            float t = 0.f;
            for (int k = 0; k < DD; ++k) t += venc[v*64 + k] * uS[hh*64 + k];
            ws[WS_TV + v*4 + hh] = t;
        }
    }
}

// ---------- K1: global per-head max of scores ----------
__global__ void k_max(const float* __restrict__ lv, const float* __restrict__ rim,
                      const int* __restrict__ idx, const float* __restrict__ attr,
                      const float* __restrict__ W1, const float* __restrict__ b1,
                      const float* __restrict__ ah, float* ws) {
    __shared__ float w1s[128], b1s[64], qs[256], aas[8], cs[4];
    __shared__ unsigned redS[4];
    int tid = threadIdx.x;
    if (tid < 128) w1s[tid] = W1[tid];
    if (tid < 64)  b1s[tid] = b1[tid];
    qs[tid] = ws[WS_Q + tid];
    if (tid < 8)   aas[tid] = ah[(tid >> 1)*130 + 128 + (tid & 1)];
    if (tid < 4) { cs[tid] = ws[WS_C + tid]; redS[tid] = enc_f(-INFINITY); }
    __syncthreads();

    int e = blockIdx.x * blockDim.x + tid;
    if (e < NE) {
        int s = idx[e], t = idx[NE + e];
        float l0 = lv[2*s], l1 = lv[2*s + 1];
        const float* R = rim + 4*t;
        float lx = l0*R[0] + l1*R[2];
        float ly = l0*R[1] + l1*R[3];
        float acc0 = cs[0], acc1 = cs[1], acc2 = cs[2], acc3 = cs[3];
        #pragma unroll 8
        for (int j = 0; j < DD; ++j) {
            float hj = fmaxf(0.f, lx*w1s[j] + ly*w1s[64 + j] + b1s[j]);
            acc0 += hj * qs[j];
            acc1 += hj * qs[64 + j];
            acc2 += hj * qs[128 + j];
            acc3 += hj * qs[192 + j];
        }
        float a0 = attr[2*e], a1 = attr[2*e + 1];
        const float* tv = ws + WS_TV + t*4;
        float sc[4] = { acc0 + tv[0] + a0*aas[0] + a1*aas[1],
                        acc1 + tv[1] + a0*aas[2] + a1*aas[3],
                        acc2 + tv[2] + a0*aas[4] + a1*aas[5],
                        acc3 + tv[3] + a0*aas[6] + a1*aas[7] };
        #pragma unroll
        for (int h = 0; h < NH; ++h) {
            float v = sc[h] > 0.f ? sc[h] : 0.01f * sc[h];
            atomicMax(&redS[h], enc_f(v));
        }
    }
    __syncthreads();
    if (tid < 4) atomicMax((unsigned*)(ws + WS_MX) + tid, redS[tid]);
}

// ---------- K2: main edge pass (WMMA f32 16x16x4) ----------
__global__ void
__launch_bounds__(256)
k_edge(const float* __restrict__ lv, const float* __restrict__ rim,
       const int* __restrict__ idx, const float* __restrict__ attr,
       const float* __restrict__ W1, const float* __restrict__ b1,
       const float* __restrict__ W2, const float* __restrict__ b2,
       const float* __restrict__ ah, float* ws) {
    __shared__ float w1s[128], b1s[64], b2s[64], qs[256], aas[8], cs[4], mxs[4];
    __shared__ float w2s[4096];
    __shared__ float hid[8][16*68];      // per-wave 16x64 hidden tile, stride 68
    __shared__ float exS[128*4];
    __shared__ int   tgtS[128];

    int tid = threadIdx.x;
    if (tid < 128) w1s[tid] = W1[tid];
    if (tid < 64)  { b1s[tid] = b1[tid]; b2s[tid] = b2[tid]; }
    qs[tid] = ws[WS_Q + tid];
    if (tid < 8)  aas[tid] = ah[(tid >> 1)*130 + 128 + (tid & 1)];
    if (tid < 4)  { cs[tid] = ws[WS_C + tid];
                    mxs[tid] = dec_f(((unsigned*)(ws + WS_MX))[tid]); }
    #pragma unroll
    for (int i = 0; i < 16; ++i) w2s[tid + 256*i] = W2[tid + 256*i];
    __syncthreads();

    float* den = ws + WS_DEN;
    float* num = ws + WS_NUM;

    // ---- phase A: per-edge gather + hidden MLP + score/exp ----
    if (tid < 128) {
        int e = blockIdx.x * 128 + tid;
        bool valid = e < NE;
        int t = 0;
        float lx = 0.f, ly = 0.f, a0 = 0.f, a1 = 0.f;
        if (valid) {
            int s = idx[e]; t = idx[NE + e];
            float l0 = lv[2*s], l1 = lv[2*s + 1];
            const float* R = rim + 4*t;
            lx = l0*R[0] + l1*R[2];
            ly = l0*R[1] + l1*R[3];
            a0 = attr[2*e]; a1 = attr[2*e + 1];
        }
        float* hrow = &hid[tid >> 4][(tid & 15) * 68];
        float acc0 = cs[0], acc1 = cs[1], acc2 = cs[2], acc3 = cs[3];
        #pragma unroll 8
        for (int j = 0; j < DD; ++j) {
            float hj = valid ? fmaxf(0.f, lx*w1s[j] + ly*w1s[64+j] + b1s[j]) : 0.f;
            hrow[j] = hj;
            acc0 += hj * qs[j];
            acc1 += hj * qs[64 + j];
            acc2 += hj * qs[128 + j];
            acc3 += hj * qs[192 + j];
        }
        const float* tv = ws + WS_TV + t*4;
        float sc[4] = { acc0 + tv[0] + a0*aas[0] + a1*aas[1],
                        acc1 + tv[1] + a0*aas[2] + a1*aas[3],
                        acc2 + tv[2] + a0*aas[4] + a1*aas[5],
                        acc3 + tv[3] + a0*aas[6] + a1*aas[7] };
        tgtS[tid] = t;
        #pragma unroll
        for (int h = 0; h < NH; ++h) {
            float v = sc[h] > 0.f ? sc[h] : 0.01f * sc[h];
            float ex = valid ? __expf(v - mxs[h]) : 0.f;
            exS[tid*4 + h] = ex;
            if (valid) atomicAdd(&den[t*4 + h], ex);
        }
    }
    __syncthreads();

    // ---- phase B: lane_enc = hidden @ W2 via V_WMMA_F32_16X16X4_F32 ----
    int wv   = tid >> 5;        // wave id -> 16-edge tile
    int lane = tid & 31;
    int hi   = lane >> 4;       // half-wave select
    int l15  = lane & 15;
    int koff = hi ? 2 : 0;

    v8f acc[4];
    #pragma unroll
    for (int n = 0; n < 4; ++n) acc[n] = (v8f){0.f,0.f,0.f,0.f,0.f,0.f,0.f,0.f};

    const float* hrow = &hid[wv][l15 * 68];
    for (int k = 0; k < DD; k += 4) {
        v2f A;
        A.x = hrow[k + koff];
        A.y = hrow[k + koff + 1];
        #pragma unroll
        for (int n = 0; n < 4; ++n) {
            v2f B;
            B.x = w2s[(k + koff    )*64 + n*16 + l15];
            B.y = w2s[(k + koff + 1)*64 + n*16 + l15];
            acc[n] = __builtin_amdgcn_wmma_f32_16x16x4_f32(
                false, A, false, B, (short)0, acc[n], false, false);
        }
    }

    // ---- phase C: weighted numerator accumulation (segment sum) ----
    #pragma unroll
    for (int n = 0; n < 4; ++n) {
        int col = n*16 + l15;
        float bb = b2s[col];
        #pragma unroll
        for (int r = 0; r < 8; ++r) {
            int M = r + (hi << 3);
            float val = acc[n][r] + bb;
            int eL = wv*16 + M;
            int tv = tgtS[eL];
            float* np = num + tv*256 + col;
            #pragma unroll
            for (int h = 0; h < NH; ++h)
                atomicAdd(np + h*64, exS[eL*4 + h] * val);
        }
    }
}

// ---------- K3: per-vehicle finalize ----------
__global__ void k_fin(const float* __restrict__ Wh, const float* __restrict__ outw,
                      const float* __restrict__ ws, float* __restrict__ out) {
    __shared__ float sld[256], hsld[256];
    int v = blockIdx.x, j = threadIdx.x;
    const float* num = ws + WS_NUM;
    const float* den = ws + WS_DEN;
    int h = j >> 6, col = j & 63;
    sld[j] = num[v*256 + j] / (den[v*4 + h] + 1e-16f);
    __syncthreads();
    float a = 0.f;
    #pragma unroll 8
    for (int i = 0; i < DD; ++i)
        a += sld[h*64 + i] * Wh[h*4096 + i*64 + col];
    hsld[j] = elu1(elu1(a));
    __syncthreads();
    if (j < DD) {
        float y = 0.f;
        #pragma unroll 8
        for (int k = 0; k < 256; ++k)
            y += hsld[k] * outw[k*64 + j];
        out[v*64 + j] = y;
    }
}

extern "C" void kernel_launch(void* const* d_in, const int* in_sizes, int n_in,
                              void* d_out, int out_size, void* d_ws, size_t ws_size,
                              hipStream_t stream) {
    const float* lv   = (const float*)d_in[0];
    const float* rim  = (const float*)d_in[1];
    const int*   idx  = (const int*)  d_in[2];
    const float* attr = (const float*)d_in[3];
    const float* venc = (const float*)d_in[4];
    const float* W1   = (const float*)d_in[5];
    const float* b1   = (const float*)d_in[6];
    const float* W2   = (const float*)d_in[7];
    const float* b2   = (const float*)d_in[8];
    const float* Wh   = (const float*)d_in[9];
    const float* ah   = (const float*)d_in[10];
    const float* outw = (const float*)d_in[11];
    float* out = (float*)d_out;
    float* ws  = (float*)d_ws;

    k_init<<<1024, 256, 0, stream>>>(ws);
    k_pre <<<1, 256, 0, stream>>>(Wh, ah, W2, b2, venc, ws);
    k_max <<<(NE + 255) / 256, 256, 0, stream>>>(lv, rim, idx, attr, W1, b1, ah, ws);
    k_edge<<<(NE + 127) / 128, 256, 0, stream>>>(lv, rim, idx, attr, W1, b1, W2, b2, ah, ws);
    k_fin <<<NV, 256, 0, stream>>>(Wh, outw, ws, out);
}